// SAGPool_66666482369198
// MI455X (gfx1250) — compile-verified
//
#include <hip/hip_runtime.h>
#include <hip/hip_bf16.h>

typedef __attribute__((ext_vector_type(2))) float v2f;
typedef __attribute__((ext_vector_type(8))) float v8f;

#define B_ 8
#define N_ 10000
#define C_ 128
#define E_ 320000
#define K_ 5000   // ceil(N * 0.5)

// workspace layout (float/uint/int elements, each region B_*N_)
#define WS_H     0
#define WS_DEG   (B_*N_)
#define WS_SCORE (2*B_*N_)
#define WS_KEY   (3*B_*N_)
#define WS_MAP   (4*B_*N_)

// ---------------------------------------------------------------------------
// 1) GEMV h = x @ W via V_WMMA_F32_16X16X4_F32.
//    One wave handles 16 rows; 32 WMMA steps accumulate over C=128 (K=4 each).
//    B-matrix columns are all broadcast-copies of W[k..k+3], so D[:,n] == h
//    for every n; lanes 0 and 16 write out rows 0-7 / 8-15 of the tile.
// ---------------------------------------------------------------------------
__global__ void k_gemv_wmma(const float* __restrict__ x,
                            const float* __restrict__ W,
                            float* __restrict__ h) {
    int wave = (int)((blockIdx.x * blockDim.x + threadIdx.x) >> 5);
    int lane = (int)(threadIdx.x & 31);
    const int nwaves = B_ * (N_ / 16);
    if (wave >= nwaves) return;
    int b  = wave / (N_ / 16);
    int rb = wave % (N_ / 16);
    int row  = rb * 16 + (lane & 15);
    int koff = (lane >> 4) * 2;                  // lanes 0-15 -> K 0,1 ; 16-31 -> K 2,3
    const float* xg = x + (size_t)b * N_ * C_ + (size_t)row * C_;

    v8f acc = {0.f, 0.f, 0.f, 0.f, 0.f, 0.f, 0.f, 0.f};
#pragma unroll
    for (int kc = 0; kc < C_ / 4; ++kc) {
        int k = kc * 4 + koff;
        v2f a;  a.x  = xg[k];  a.y  = xg[k + 1];   // A: 16x4 fragment
        v2f bb; bb.x = W[k];   bb.y = W[k + 1];    // B: 4x16, every column = W chunk
        acc = __builtin_amdgcn_wmma_f32_16x16x4_f32(
            /*neg_a=*/false, a, /*neg_b=*/false, bb,
            /*c_mod=*/(short)0, acc, /*reuse_a=*/false, /*reuse_b=*/false);
    }
    // C/D layout: lanes 0-15 hold M=v (VGPR v), lanes 16-31 hold M=v+8.
    if ((lane & 15) == 0) {
        int mbase = rb * 16 + ((lane >> 4) ? 8 : 0);
        float* hg = h + (size_t)b * N_;
#pragma unroll
        for (int v = 0; v < 8; ++v) hg[mbase + v] = acc[v];
    }
}

// ---------------------------------------------------------------------------
// 2) zero deg + score accumulators
// ---------------------------------------------------------------------------
__global__ void k_zero(float* __restrict__ ws) {
    int i = (int)(blockIdx.x * blockDim.x + threadIdx.x);
    if (i < 2 * B_ * N_) ws[WS_DEG + i] = 0.0f;
}

// 3) degree: atomic count of incoming edges
__global__ void k_deg(const int* __restrict__ ei, float* __restrict__ deg) {
    int i = (int)(blockIdx.x * blockDim.x + threadIdx.x);
    if (i >= B_ * E_) return;
    int b = i / E_, e = i - b * E_;
    int dst = ei[(size_t)b * 2 * E_ + E_ + e];
    atomicAdd(&deg[b * N_ + dst], 1.0f);
}

// 4) dinv = rsqrt(deg + 1)   (in place)
__global__ void k_dinv(float* __restrict__ deg) {
    int i = (int)(blockIdx.x * blockDim.x + threadIdx.x);
    if (i < B_ * N_) deg[i] = rsqrtf(deg[i] + 1.0f);
}

// 5) edge aggregation: score[dst] += h[src] * dinv[src] * dinv[dst]
__global__ void k_agg(const int* __restrict__ ei, const float* __restrict__ h,
                      const float* __restrict__ dinv, float* __restrict__ score) {
    int i = (int)(blockIdx.x * blockDim.x + threadIdx.x);
    if (i >= B_ * E_) return;
    int b = i / E_, e = i - b * E_;
    const int* eb = ei + (size_t)b * 2 * E_;
    int src = eb[e], dst = eb[E_ + e];
    float c = dinv[b * N_ + src] * dinv[b * N_ + dst];
    atomicAdd(&score[b * N_ + dst], h[b * N_ + src] * c);
}

// 6) finalize score (self-loop + bias) and build order-preserving uint key
__global__ void k_keys(const float* __restrict__ h, const float* __restrict__ dinv,
                       const float* __restrict__ score, const float* __restrict__ bias,
                       unsigned* __restrict__ keys) {
    int i = (int)(blockIdx.x * blockDim.x + threadIdx.x);
    if (i >= B_ * N_) return;
    float s = score[i] + h[i] * dinv[i] * dinv[i] + bias[0];
    unsigned u = __float_as_uint(s);
    keys[i] = (u & 0x80000000u) ? ~u : (u | 0x80000000u);
}

// ---------------------------------------------------------------------------
// 7) per-graph top-K selection + compaction map. One 1024-thread WG per graph.
//    LDS: 4096-bin histogram of key[31:20] (16KB) + 1024-entry scan buffer.
// ---------------------------------------------------------------------------
__device__ __forceinline__ unsigned scan_excl(unsigned v, unsigned* sbuf,
                                              int tid, unsigned* total) {
    sbuf[tid] = v;
    __syncthreads();
    for (int off = 1; off < 1024; off <<= 1) {
        unsigned t = (tid >= off) ? sbuf[tid - off] : 0u;
        __syncthreads();
        sbuf[tid] += t;
        __syncthreads();
    }
    unsigned incl = sbuf[tid];
    *total = sbuf[1023];
    __syncthreads();
    return incl - v;
}

__global__ void __launch_bounds__(1024)
k_select(const unsigned* __restrict__ keys, int* __restrict__ mapping) {
    __shared__ unsigned hist[4096];
    __shared__ unsigned sbuf[1024];
    __shared__ int sT;
    __shared__ unsigned sR, sKept, sTSeen;

    int b   = (int)blockIdx.x;
    int tid = (int)threadIdx.x;
    const unsigned* kg = keys + (size_t)b * N_;

    for (int i = tid; i < 4096; i += 1024) hist[i] = 0u;
    __syncthreads();
    for (int n = tid; n < N_; n += 1024) atomicAdd(&hist[kg[n] >> 20], 1u);
    __syncthreads();

    if (tid == 0) {
        unsigned cum = 0; int T = 0; unsigned R = K_;
        for (int bin = 4095; bin >= 0; --bin) {
            unsigned c = hist[bin];
            if (cum + c >= (unsigned)K_) { T = bin; R = (unsigned)K_ - cum; break; }
            cum += c;
        }
        sT = T; sR = R; sKept = 0u; sTSeen = 0u;
    }
    __syncthreads();
    unsigned T = (unsigned)sT;

    for (int base = 0; base < N_; base += 1024) {
        int n = base + tid;
        int valid = (n < N_);
        unsigned bin    = valid ? (kg[n] >> 20) : 0u;
        unsigned isT    = (valid && bin == T) ? 1u : 0u;
        unsigned keepHi = (valid && bin >  T) ? 1u : 0u;

        unsigned totT;
        unsigned exT = scan_excl(isT, sbuf, tid, &totT);
        unsigned keep = keepHi | ((isT && (sTSeen + exT < sR)) ? 1u : 0u);

        unsigned totK;
        unsigned exK = scan_excl(keep, sbuf, tid, &totK);
        if (valid) mapping[(size_t)b * N_ + n] = keep ? (int)(sKept + exK) : -1;
        __syncthreads();
        if (tid == 0) { sKept += totK; sTSeen += totT; }
        __syncthreads();
    }
}

// 8) gather kept rows: one wave per node, one float4 per lane (128 floats)
__global__ void k_gather(const float* __restrict__ x, const int* __restrict__ mapping,
                         float* __restrict__ xpool) {
    int wave = (int)((blockIdx.x * blockDim.x + threadIdx.x) >> 5);
    int lane = (int)(threadIdx.x & 31);
    if (wave >= B_ * N_) return;
    int b = wave / N_, n = wave - b * N_;
    int pos = mapping[(size_t)b * N_ + n];
    if (pos < 0) return;
    const float4* src = (const float4*)(x + ((size_t)b * N_ + n) * C_);
    float4*       dst = (float4*)(xpool + ((size_t)b * K_ + pos) * C_);
    dst[lane] = src[lane];
}

// 9) edge filter + remap + mask
__global__ void k_edges(const int* __restrict__ ei, const int* __restrict__ mapping,
                        int* __restrict__ outEI, float* __restrict__ outMask) {
    int i = (int)(blockIdx.x * blockDim.x + threadIdx.x);
    if (i >= B_ * E_) return;
    int b = i / E_, e = i - b * E_;
    const int* eb = ei + (size_t)b * 2 * E_;
    int ns = mapping[(size_t)b * N_ + eb[e]];
    int nd = mapping[(size_t)b * N_ + eb[E_ + e]];
    bool keep = (ns >= 0) && (nd >= 0);
    int* ob = outEI + (size_t)b * 2 * E_;
    ob[e]       = keep ? ns : -1;
    ob[E_ + e]  = keep ? nd : -1;
    outMask[(size_t)b * E_ + e] = keep ? 1.0f : 0.0f;
}

// ---------------------------------------------------------------------------
extern "C" void kernel_launch(void* const* d_in, const int* in_sizes, int n_in,
                              void* d_out, int out_size, void* d_ws, size_t ws_size,
                              hipStream_t stream) {
    const float* x    = (const float*)d_in[0];   // [B, N, C]
    const int*   ei   = (const int*)  d_in[1];   // [B, 2, E]
    const float* W    = (const float*)d_in[2];   // [C, 1]
    const float* bias = (const float*)d_in[3];   // [1]

    float*    ws      = (float*)d_ws;
    float*    h       = ws + WS_H;
    float*    deg     = ws + WS_DEG;     // becomes dinv in-place
    float*    score   = ws + WS_SCORE;
    unsigned* keys    = (unsigned*)(ws + WS_KEY);
    int*      mapping = (int*)(ws + WS_MAP);

    float* out     = (float*)d_out;
    float* xpool   = out;                                  // B*K*C floats
    int*   outEI   = (int*)(out + (size_t)B_ * K_ * C_);   // B*2*E ints
    float* outMask = out + (size_t)B_ * K_ * C_ + (size_t)B_ * 2 * E_; // B*E

    const int TB = 256;
    k_zero     <<<(2 * B_ * N_ + TB - 1) / TB, TB, 0, stream>>>(ws);
    k_gemv_wmma<<<(B_ * (N_ / 16) * 32 + TB - 1) / TB, TB, 0, stream>>>(x, W, h);
    k_deg      <<<(B_ * E_ + TB - 1) / TB, TB, 0, stream>>>(ei, deg);
    k_dinv     <<<(B_ * N_ + TB - 1) / TB, TB, 0, stream>>>(deg);
    k_agg      <<<(B_ * E_ + TB - 1) / TB, TB, 0, stream>>>(ei, h, deg, score);
    k_keys     <<<(B_ * N_ + TB - 1) / TB, TB, 0, stream>>>(h, deg, score, bias, keys);
    k_select   <<<B_, 1024, 0, stream>>>(keys, mapping);
    k_gather   <<<(B_ * N_ * 32 + TB - 1) / TB, TB, 0, stream>>>(x, mapping, xpool);
    k_edges    <<<(B_ * E_ + TB - 1) / TB, TB, 0, stream>>>(ei, mapping, outEI, outMask);
}